// DNeRFNGP_40681930227972
// MI455X (gfx1250) — compile-verified
//
#include <hip/hip_runtime.h>
#include <cmath>

// ---------------------------------------------------------------------------
// D-NeRF NGP encoder for gfx1250 (MI455X).
//   Stage 1 (delta): 16 points per wave32. Each lane computes plane-products
//     for (point = lane%16, 2 channels selected by lane/16), and the 64-channel
//     sum is accumulated with V_WMMA_F32_16X16X4_F32 (B = ones) in the XDL
//     pipe while VALU does the bilinear math.
//   Stage 2 (hash): per point, 32 lanes cover 16 levels x 8 corners in 4
//     passes; 8-lane shuffle-xor tree reduces the corner sum; float2 stores.
// Everything is L2-resident (feats 59MB + table 49MB < 192MB L2), so the
// kernel is gather-latency bound; all gathers are b64 (x0,x0+1 pairs).
// ---------------------------------------------------------------------------

typedef float v2f __attribute__((ext_vector_type(2)));
typedef float v8f __attribute__((ext_vector_type(8)));

struct HashParams {
  int scales[16];
  int offsets[16];
  int start_hash;
  unsigned int n_entries;
  double inv_n;
};

#define WAVES_PER_BLOCK 8
#define PTS_PER_WAVE 16

__device__ __forceinline__ float2 ld2(const float* p) {
  float2 v;
  __builtin_memcpy(&v, p, sizeof(float2));  // align-4 b64 load (hw supports)
  return v;
}

__device__ __forceinline__ v8f wmma_rowsum4(v2f a, v2f b, v8f c) {
#if __has_builtin(__builtin_amdgcn_wmma_f32_16x16x4_f32)
  // D = A(16x4) * B(4x16) + C ; with B == ones: D[m,n] = sum_k A[m,k] + C
  return __builtin_amdgcn_wmma_f32_16x16x4_f32(false, a, false, b, (short)0, c,
                                               false, false);
#else
  asm volatile("v_wmma_f32_16x16x4_f32 %0, %1, %2, %0"
               : "+v"(c)
               : "v"(a), "v"(b));
  return c;
#endif
}

__global__ __launch_bounds__(256) void dnerf_ngp_kernel(
    const float* __restrict__ xin, const float* __restrict__ wbounds,
    const float* __restrict__ table, const float* __restrict__ f0,
    const float* __restrict__ f1, const float* __restrict__ f2,
    float* __restrict__ out, HashParams P) {
  __shared__ float s_scalef[16];
  __shared__ int s_s1[16];
  __shared__ int s_off[16];
  __shared__ float s_enc[WAVES_PER_BLOCK][PTS_PER_WAVE][3];
  __shared__ float s_dst[WAVES_PER_BLOCK][PTS_PER_WAVE][3];

  const int tid = threadIdx.x;
  if (tid == 0) {
#pragma unroll
    for (int i = 0; i < 16; ++i) {
      s_scalef[i] = (float)P.scales[i];
      s_s1[i] = P.scales[i] + 1;
      s_off[i] = P.offsets[i];
    }
  }
  __syncthreads();

  const int wave = tid >> 5;
  const int lane = tid & 31;
  const int m = lane & 15;   // point within wave (WMMA row M)
  const int half = lane >> 4;  // K half: lanes 0-15 -> K{0,1}, 16-31 -> K{2,3}
  const int base = (blockIdx.x * WAVES_PER_BLOCK + wave) * PTS_PER_WAVE;
  const int n = base + m;

  // ---- normalized inputs (matches clip/sub/div order of reference) --------
  const float lo0 = wbounds[0], lo1 = wbounds[1], lo2 = wbounds[2];
  const float hi0 = wbounds[3], hi1 = wbounds[4], hi2 = wbounds[5];
  const float mx = fmaxf(hi0 - lo0, fmaxf(hi1 - lo1, hi2 - lo2));

  const float px = xin[n * 4 + 0], py = xin[n * 4 + 1];
  const float pz = xin[n * 4 + 2], ptv = xin[n * 4 + 3];
  const float in0 = (fminf(fmaxf(px, lo0), hi0) - lo0) / mx;
  const float in1 = (fminf(fmaxf(py, lo1), hi1) - lo1) / mx;
  const float in2 = (fminf(fmaxf(pz, lo2), hi2) - lo2) / mx;
  const float tn = ptv / 99.0f;  // t / (NUM_FRAMES-1)

  // ---- grid-sample coordinates (exact f32 op sequence of reference) -------
  float gx[3];
  gx[0] = ((2.0f * in0 - 1.0f) + 1.0f) * 0.5f * 255.0f;
  gx[1] = ((2.0f * in1 - 1.0f) + 1.0f) * 0.5f * 255.0f;
  gx[2] = ((2.0f * in2 - 1.0f) + 1.0f) * 0.5f * 255.0f;
  float gy = ((2.0f * tn - 1.0f) + 1.0f) * 0.5f * 99.0f;

  int iy0 = (int)floorf(gy);
  if (iy0 > 98) iy0 = 98;  // boundary trick == reference zero-padding
  const float wy = gy - (float)iy0;
  const float wya = 1.0f - wy;

  int jOff[3];
  float W00[3], W01[3], W10[3], W11[3];
#pragma unroll
  for (int j = 0; j < 3; ++j) {
    int ix0 = (int)floorf(gx[j]);
    if (ix0 > 254) ix0 = 254;
    const float wx = gx[j] - (float)ix0;
    const float wxa = 1.0f - wx;
    W00[j] = wxa * wya;
    W01[j] = wx * wya;
    W10[j] = wxa * wy;
    W11[j] = wx * wy;
    jOff[j] = j * 1638400 + iy0 * 256 + ix0;  // j*(64*100*256) + y*256 + x
  }

  // ---- stage 1: delta via WMMA channel-sum accumulation -------------------
  v8f D0 = {0, 0, 0, 0, 0, 0, 0, 0};
  v8f D1 = {0, 0, 0, 0, 0, 0, 0, 0};
  v8f D2 = {0, 0, 0, 0, 0, 0, 0, 0};
  const v2f Bones = {1.0f, 1.0f};

  for (int t4 = 0; t4 < 16; ++t4) {  // 16 K-chunks of 4 channels
    float pA[3][2];
#pragma unroll
    for (int cc = 0; cc < 2; ++cc) {
      const int c = (t4 << 2) + (half << 1) + cc;
      const int cOff = c * 25600;  // channel stride = 100*256
      float q0 = 1.0f, q1 = 1.0f, q2 = 1.0f;
#pragma unroll
      for (int j = 0; j < 3; ++j) {
        const int o = jOff[j] + cOff;
        const float2 a0 = ld2(f0 + o), b0 = ld2(f0 + o + 256);
        const float2 a1 = ld2(f1 + o), b1 = ld2(f1 + o + 256);
        const float2 a2 = ld2(f2 + o), b2 = ld2(f2 + o + 256);
        const float c00 = W00[j], c01 = W01[j], c10 = W10[j], c11 = W11[j];
        float s0 = fmaf(a0.y, c01, a0.x * c00);
        s0 = fmaf(b0.x, c10, s0);
        s0 = fmaf(b0.y, c11, s0);
        float s1 = fmaf(a1.y, c01, a1.x * c00);
        s1 = fmaf(b1.x, c10, s1);
        s1 = fmaf(b1.y, c11, s1);
        float s2 = fmaf(a2.y, c01, a2.x * c00);
        s2 = fmaf(b2.x, c10, s2);
        s2 = fmaf(b2.y, c11, s2);
        q0 *= s0;
        q1 *= s1;
        q2 *= s2;
      }
      pA[0][cc] = q0;
      pA[1][cc] = q1;
      pA[2][cc] = q2;
    }
    v2f A0 = {pA[0][0], pA[0][1]};
    v2f A1 = {pA[1][0], pA[1][1]};
    v2f A2 = {pA[2][0], pA[2][1]};
    D0 = wmma_rowsum4(A0, Bones, D0);
    D1 = wmma_rowsum4(A1, Bones, D1);
    D2 = wmma_rowsum4(A2, Bones, D2);
  }

  // D[m, n] identical in every column n. Lanes<16 hold points 0-7 in
  // VGPRs 0-7; lanes>=16 hold points 8-15. Stage via LDS.
  if (lane == 0) {
#pragma unroll
    for (int r = 0; r < 8; ++r) {
      s_dst[wave][r][0] = D0[r];
      s_dst[wave][r][1] = D1[r];
      s_dst[wave][r][2] = D2[r];
    }
  }
  if (lane == 16) {
#pragma unroll
    for (int r = 0; r < 8; ++r) {
      s_dst[wave][8 + r][0] = D0[r];
      s_dst[wave][8 + r][1] = D1[r];
      s_dst[wave][8 + r][2] = D2[r];
    }
  }
  __syncthreads();

  const float ENC_MAX = (float)(1.0 - 1e-6);
  if (lane < 16) {  // lane m owns point m's normalized inputs already
    s_enc[wave][m][0] = fminf(fmaxf(in0 + s_dst[wave][m][0], 0.0f), ENC_MAX);
    s_enc[wave][m][1] = fminf(fmaxf(in1 + s_dst[wave][m][1], 0.0f), ENC_MAX);
    s_enc[wave][m][2] = fminf(fmaxf(in2 + s_dst[wave][m][2], 0.0f), ENC_MAX);
  }
  __syncthreads();

  // ---- stage 2: multiresolution hash encoding -----------------------------
  const int corner = lane & 7;
  const int lgrp = lane >> 3;  // 4 level-groups of 8 corner lanes
  const int cx = (corner >> 2) & 1;
  const int cy = (corner >> 1) & 1;
  const int cz = corner & 1;

  for (int mm = 0; mm < 16; ++mm) {
    const float e0 = s_enc[wave][mm][0];
    const float e1 = s_enc[wave][mm][1];
    const float e2 = s_enc[wave][mm][2];
    const int outn = base + mm;
#pragma unroll
    for (int k = 0; k < 4; ++k) {
      const int l = lgrp + 4 * k;
      const float sf = s_scalef[l];
      const int s1 = s_s1[l];
      const int off = s_off[l];
      const float fx0 = e0 * sf, fx1 = e1 * sf, fx2 = e2 * sf;
      const float fl0 = floorf(fx0), fl1 = floorf(fx1), fl2 = floorf(fx2);
      const float o0 = fx0 - fl0, o1 = fx1 - fl1, o2 = fx2 - fl2;
      const int i0 = (int)fl0 + cx;
      const int i1 = (int)fl1 + cy;
      const int i2 = (int)fl2 + cz;

      long long ind;
      if (l >= P.start_hash) {
        const unsigned long long h = (unsigned long long)(unsigned)i0 * 1ull ^
                                     (unsigned long long)(unsigned)i1 * 19349663ull ^
                                     (unsigned long long)(unsigned)i2 * 83492791ull;
        // h < 2^38: double-reciprocal mod with +-1 fixup is exact
        const double qd = (double)h * P.inv_n;
        long long q = (long long)qd;
        long long r = (long long)h - q * (long long)P.n_entries;
        if (r < 0) r += (long long)P.n_entries;
        else if (r >= (long long)P.n_entries) r -= (long long)P.n_entries;
        ind = r + off;
      } else {
        ind = (long long)(i0 * (s1 * s1) + i1 * s1 + i2) + off;
      }

      const float2 val = ld2(table + 2 * ind);
      const float wa = cx ? o0 : 1.0f - o0;
      const float wb = cy ? o1 : 1.0f - o1;
      const float wc = cz ? o2 : 1.0f - o2;
      const float w = wa * wb * wc;
      float r0 = w * val.x;
      float r1 = w * val.y;
      // 8-corner reduction within each level group
      r0 += __shfl_xor(r0, 1);
      r1 += __shfl_xor(r1, 1);
      r0 += __shfl_xor(r0, 2);
      r1 += __shfl_xor(r1, 2);
      r0 += __shfl_xor(r0, 4);
      r1 += __shfl_xor(r1, 4);
      if (corner == 0) {
        float2 res;
        res.x = r0;
        res.y = r1;
        *(float2*)(out + (long long)outn * 32 + 2 * l) = res;
      }
    }
  }
}

// ---------------------------------------------------------------------------
static bool isprime_h(long long v) {
  if (v < 2) return false;
  for (long long i = 2; i * i <= v; ++i)
    if (v % i == 0) return false;
  return true;
}

extern "C" void kernel_launch(void* const* d_in, const int* in_sizes, int n_in,
                              void* d_out, int out_size, void* d_ws,
                              size_t ws_size, hipStream_t stream) {
  (void)n_in;
  (void)d_ws;
  (void)ws_size;
  (void)out_size;
  const float* x = (const float*)d_in[0];
  const float* wb = (const float*)d_in[1];
  const float* table = (const float*)d_in[2];
  const float* f0 = (const float*)d_in[3];
  const float* f1 = (const float*)d_in[4];
  const float* f2 = (const float*)d_in[5];
  float* out = (float*)d_out;

  // Derive constants with the same libm pow() CPython uses -> bit-identical
  // SCALES (level 15 sits at 2048 +- ~1e-12, so this must match the host).
  HashParams P;
  long long n_entries = 1ll << 19;  // LOG2_HASHMAP = 19
  while (!isprime_h(n_entries)) ++n_entries;
  const double b = pow(2048.0 / 16.0, 1.0 / 15.0);
  long long offs = 0;
  int start_hash = -1;
  for (int i = 0; i < 16; ++i) {
    const int res = (int)(16.0 * pow(b, (double)i));
    P.scales[i] = res;
    P.offsets[i] = (int)offs;
    long long ne = (long long)(res + 1) * (res + 1) * (res + 1);
    if (ne > n_entries) {
      if (start_hash < 0) start_hash = i;
      ne = n_entries;
    }
    offs += ne;
  }
  P.start_hash = (start_hash < 0) ? 16 : start_hash;
  P.n_entries = (unsigned)n_entries;
  P.inv_n = 1.0 / (double)n_entries;

  const int npts = in_sizes[0] / 4;                      // 131072
  const int blocks = npts / (WAVES_PER_BLOCK * PTS_PER_WAVE);  // 1024
  dnerf_ngp_kernel<<<blocks, 256, 0, stream>>>(x, wb, table, f0, f1, f2, out,
                                               P);
}